// S2Convolution_78554951844156
// MI455X (gfx1250) — compile-verified
//
#include <hip/hip_runtime.h>

// S2Convolution for MI455X (gfx1250, wave32).
// out[n,ab,o,l,k] = sum_{i<5} x[n,ab,i,l] * w[ab,i,o,k]   (+bias at degree 0)
// Viewed per (ab,l) as GEMM: C[n, c] = sum_i x[n,ab,i,l] * w[ab,i,c], c = o*M+k,
// M(rows)=4096, N(cols)=8*M, K=5 -> V_WMMA_F32_16X16X4_F32 x2 (K=4 + padded K=1).
// Store-bandwidth bound (666 MB out vs 33 MB in) -> nontemporal coalesced stores.

typedef float v2f __attribute__((ext_vector_type(2)));
typedef float v8f __attribute__((ext_vector_type(8)));

#define NROWS 4096
#define AB    9
#define BIN   5
#define BOUT  8

template <int M>
__global__ __launch_bounds__(32)
void s2conv_wmma_kernel(const float* __restrict__ x,   // (N, AB, BIN, M)
                        const float* __restrict__ w,   // (AB, BIN, BOUT, M)
                        const float* __restrict__ bias,// (AB, BOUT) flat, only for M==1
                        float* __restrict__ out)       // (N, AB, BOUT, M, M)
{
    constexpr int NC = BOUT * M;          // GEMM column count (o,k flattened)
    constexpr int CT = (NC + 15) / 16;    // 16-wide column tiles

    const int lane = threadIdx.x & 31;
    const int l16  = lane & 15;           // column within tile / row-of-A lane
    const int half = lane >> 4;           // K-slot selector per ISA A/B layout

    const int ab = blockIdx.x;            // 0..8
    const int n0 = blockIdx.y * 16;       // n-tile base, gridDim.y = NROWS/16

    const long xN  = (long)AB * BIN * M;              // x stride over n
    const long oN  = (long)AB * BOUT * M * M;         // out stride over n
    const float* xab = x   + (long)ab * BIN * M;
    const float* wab = w   + (long)ab * BIN * NC;     // w[ab,i,c], c contiguous
    float*       oab = out + (long)ab * BOUT * M * M;

    // ---- B tiles: load once per block, keep in registers across l / stores ----
    // B(4x16) layout: vgpr v, lanes0-15 -> K=v, lanes16-31 -> K=v+2, N=lane%16.
    v2f b0[CT];   // K rows i=0..3
    v2f b1[CT];   // K row  i=4 placed in K-slot 0, rest zero
    #pragma unroll
    for (int ct = 0; ct < CT; ++ct) {
        const int  c  = ct * 16 + l16;
        const bool ok = (c < NC);
        b0[ct].x = ok ? wab[(long)(2 * half + 0) * NC + c] : 0.0f;
        b0[ct].y = ok ? wab[(long)(2 * half + 1) * NC + c] : 0.0f;
        b1[ct].x = (ok && half == 0) ? wab[(long)4 * NC + c] : 0.0f;
        b1[ct].y = 0.0f;
    }

    for (int l = 0; l < M; ++l) {
        // ---- A tile (16 rows x K): lane = row, K = vgpr + 2*half ----
        const float* xrow = xab + (long)(n0 + l16) * xN + l;
        v2f a0, a1;
        a0.x = xrow[(long)(2 * half + 0) * M];
        a0.y = xrow[(long)(2 * half + 1) * M];
        a1.x = (half == 0) ? xrow[(long)4 * M] : 0.0f;  // i=4 in K-slot 0
        a1.y = 0.0f;

        #pragma unroll
        for (int ct = 0; ct < CT; ++ct) {
            const int c = ct * 16 + l16;

            v8f acc;
            if (M == 1) {
                // degree 0: seed accumulator with bias[ab,o] (same for every row)
                const float bv = (c < NC) ? bias[ab * BOUT + c] : 0.0f;
                acc = (v8f){bv, bv, bv, bv, bv, bv, bv, bv};
            } else {
                acc = (v8f){0.f, 0.f, 0.f, 0.f, 0.f, 0.f, 0.f, 0.f};
            }

            // D = A*B + C, f32 16x16x4, two K-blocks covering K=5
            acc = __builtin_amdgcn_wmma_f32_16x16x4_f32(
                      false, a0, false, b0[ct], (short)0, acc, false, false);
            acc = __builtin_amdgcn_wmma_f32_16x16x4_f32(
                      false, a1, false, b1[ct], (short)0, acc, false, false);

            // ---- store: C vgpr r -> row (r + 8*half), col = l16 ----
            if (c < NC) {
                const int  o    = c / M;       // compile-time M -> strength-reduced
                const int  k    = c % M;
                const long coff = (long)o * M * M + (long)l * M + k;
                #pragma unroll
                for (int r = 0; r < 8; ++r) {
                    const int nl = r + 8 * half;
                    __builtin_nontemporal_store(
                        acc[r], &oab[(long)(n0 + nl) * oN + coff]);
                }
            }
        }
    }
}

extern "C" void kernel_launch(void* const* d_in, const int* in_sizes, int n_in,
                              void* d_out, int out_size, void* d_ws, size_t ws_size,
                              hipStream_t stream) {
    // setup_inputs order: x0,w0, x2,w2, x4,w4, x6,w6, x8,w8, bias
    const float* x0 = (const float*)d_in[0];
    const float* w0 = (const float*)d_in[1];
    const float* x2 = (const float*)d_in[2];
    const float* w2 = (const float*)d_in[3];
    const float* x4 = (const float*)d_in[4];
    const float* w4 = (const float*)d_in[5];
    const float* x6 = (const float*)d_in[6];
    const float* w6 = (const float*)d_in[7];
    const float* x8 = (const float*)d_in[8];
    const float* w8 = (const float*)d_in[9];
    const float* bias = (const float*)d_in[10];
    float* out = (float*)d_out;

    // output tuple offsets (floats): N*AB*BOUT*m^2 each
    const long off0 = 0;
    const long off2 = off0 + (long)NROWS * AB * BOUT * 1 * 1;    //   294912
    const long off4 = off2 + (long)NROWS * AB * BOUT * 5 * 5;    //  7667712
    const long off6 = off4 + (long)NROWS * AB * BOUT * 9 * 9;    // 31555584
    const long off8 = off6 + (long)NROWS * AB * BOUT * 13 * 13;  // 81395712

    dim3 grid(AB, NROWS / 16);   // 9 x 256 waves per degree
    dim3 block(32);              // one wave32 per block

    s2conv_wmma_kernel<1 ><<<grid, block, 0, stream>>>(x0, w0, bias, out + off0);
    s2conv_wmma_kernel<5 ><<<grid, block, 0, stream>>>(x2, w2, nullptr, out + off2);
    s2conv_wmma_kernel<9 ><<<grid, block, 0, stream>>>(x4, w4, nullptr, out + off4);
    s2conv_wmma_kernel<13><<<grid, block, 0, stream>>>(x6, w6, nullptr, out + off6);
    s2conv_wmma_kernel<17><<<grid, block, 0, stream>>>(x8, w8, nullptr, out + off8);
}